// FocalClassBalancedLoss_52853867545034
// MI455X (gfx1250) — compile-verified
//
#include <hip/hip_runtime.h>
#include <math.h>

// FocalClassBalancedLoss for MI455X (gfx1250, wave32).
// Memory-bound streaming reduction; WMMA f32 16x16x4 (B = ones) used as the
// in-wave accumulation tree on the matrix pipe, co-executing with VALU.

typedef float v2f __attribute__((ext_vector_type(2)));
typedef float v8f __attribute__((ext_vector_type(8)));

#define FCB_ALPHA 0.5f
#define FCB_BETA  0.9999f
#define FCB_C     1024

// ---------------------------------------------------------------- init ws
__global__ void fcb_init(float* __restrict__ ws) {
    if (threadIdx.x == 0) { ws[0] = 0.0f; ws[1] = 0.0f; }
}

// ------------------------------------------------- block reduce (8 waves)
// WMMA accumulator D[m][n] has identical columns; lane 0 holds D[0..7][0],
// lane 16 holds D[8..15][0]. Sum those two lanes' 8-component sums.
__device__ __forceinline__ void fcb_block_commit(v8f acc, float tail,
                                                 float* __restrict__ gaccum,
                                                 float* __restrict__ red /*16 floats LDS*/) {
    float s = acc[0] + acc[1] + acc[2] + acc[3] + acc[4] + acc[5] + acc[6] + acc[7];
    const int lane = threadIdx.x & 31;
    const int wave = threadIdx.x >> 5;
    if (lane == 0)       red[2 * wave]     = s;
    else if (lane == 16) red[2 * wave + 1] = s;
    __syncthreads();
    if (threadIdx.x == 0) {
        float tot = 0.0f;
#pragma unroll
        for (int i = 0; i < 16; ++i) tot += red[i];
        atomicAdd(gaccum, tot);
    }
    if (tail != 0.0f) atomicAdd(gaccum, tail);  // in practice no tail exists
}

// ------------------------------------------------ pass 1: count of ones
__global__ __launch_bounds__(256) void fcb_count(const float4* __restrict__ tg4,
                                                 float* __restrict__ ws, int n4) {
    const int tid    = blockIdx.x * blockDim.x + threadIdx.x;
    const int stride = gridDim.x * blockDim.x;
    const int iters  = n4 / stride;          // uniform trip count, EXEC all-1s

    v8f acc = {};
    v2f bones; bones[0] = 1.0f; bones[1] = 1.0f;

    for (int it = 0; it < iters; ++it) {
        const int i4 = tid + it * stride;
        __builtin_prefetch((const void*)(tg4 + i4 + stride), 0, 0);
        float4 t = tg4[i4];                  // global_load_b128
        v2f a; a[0] = t.x + t.y; a[1] = t.z + t.w;
        acc = __builtin_amdgcn_wmma_f32_16x16x4_f32(false, a, false, bones,
                                                    (short)0, acc, false, false);
    }
    float tail = 0.0f;
    const int i4 = tid + iters * stride;
    if (i4 < n4) { float4 t = tg4[i4]; tail = t.x + t.y + t.z + t.w; }

    __shared__ float red[16];
    fcb_block_commit(acc, tail, &ws[0], red);
}

// ----------------------------------------------- per-element focal loss
__device__ __forceinline__ float fcb_loss(float x, float t, float w) {
    float ax  = fabsf(x);
    float bce = fmaxf(x, 0.0f) - x * t + log1pf(__expf(-ax));
    float pt  = __expf(-bce);
    float om  = 1.0f - pt;
    return w * (FCB_ALPHA * om * om * bce);
}

// ------------------------------------------------ pass 2: weighted loss
__global__ __launch_bounds__(256) void fcb_lossred(const float4* __restrict__ in4,
                                                   const float4* __restrict__ tg4,
                                                   float* __restrict__ ws,
                                                   int n4, float total_elems) {
    // Class-balanced weights from global one-count (3 distinct values).
    const float cnt1 = ws[0];
    const float cnt0 = total_elems - cnt1;
    const float e0   = (cnt0 == 0.0f) ? 1e-6f : (1.0f - __powf(FCB_BETA, cnt0));
    const float e1   = (cnt1 == 0.0f) ? 1e-6f : (1.0f - __powf(FCB_BETA, cnt1));
    const float w0r  = (1.0f - FCB_BETA) / e0;
    const float w1r  = (1.0f - FCB_BETA) / e1;
    const float wrr  = (1.0f - FCB_BETA) / 1e-6f;            // classes 2..C-1
    const float sumw = w0r + w1r + (float)(FCB_C - 2) * wrr;
    const float scl  = (float)FCB_C / sumw;
    const float w0 = w0r * scl, w1 = w1r * scl, wr = wrr * scl;

    const int tid    = blockIdx.x * blockDim.x + threadIdx.x;
    const int stride = gridDim.x * blockDim.x;
    const int iters  = n4 / stride;

    v8f acc = {};
    v2f bones; bones[0] = 1.0f; bones[1] = 1.0f;

    for (int it = 0; it < iters; ++it) {
        const int i4 = tid + it * stride;
        __builtin_prefetch((const void*)(in4 + i4 + stride), 0, 0);
        __builtin_prefetch((const void*)(tg4 + i4 + stride), 0, 0);
        float4 x = in4[i4];
        float4 t = tg4[i4];
        // column of element j of this float4 is ((i4*4 + j) & 1023); within a
        // float4 no wrap occurs, so only j=0,1 at a row start differ.
        const bool at_start = ((i4 & (FCB_C / 4 - 1)) == 0);
        const float wA = at_start ? w0 : wr;
        const float wB = at_start ? w1 : wr;
        float l0 = fcb_loss(x.x, t.x, wA);
        float l1 = fcb_loss(x.y, t.y, wB);
        float l2 = fcb_loss(x.z, t.z, wr);
        float l3 = fcb_loss(x.w, t.w, wr);
        v2f a; a[0] = l0 + l1; a[1] = l2 + l3;
        acc = __builtin_amdgcn_wmma_f32_16x16x4_f32(false, a, false, bones,
                                                    (short)0, acc, false, false);
    }
    float tail = 0.0f;
    const int i4 = tid + iters * stride;
    if (i4 < n4) {
        float4 x = in4[i4];
        float4 t = tg4[i4];
        const bool at_start = ((i4 & (FCB_C / 4 - 1)) == 0);
        tail = fcb_loss(x.x, t.x, at_start ? w0 : wr) +
               fcb_loss(x.y, t.y, at_start ? w1 : wr) +
               fcb_loss(x.z, t.z, wr) +
               fcb_loss(x.w, t.w, wr);
    }

    __shared__ float red[16];
    fcb_block_commit(acc, tail, &ws[1], red);
}

// ---------------------------------------------------------------- finalize
__global__ void fcb_finalize(const float* __restrict__ ws, float* __restrict__ out,
                             float inv_total) {
    if (threadIdx.x == 0) out[0] = ws[1] * inv_total;
}

// ---------------------------------------------------------------- launcher
extern "C" void kernel_launch(void* const* d_in, const int* in_sizes, int n_in,
                              void* d_out, int out_size, void* d_ws, size_t ws_size,
                              hipStream_t stream) {
    const float* inputs  = (const float*)d_in[0];
    const float* targets = (const float*)d_in[1];
    float*       out     = (float*)d_out;
    float*       ws      = (float*)d_ws;

    const int n_total = in_sizes[0];          // 16384 * 1024 = 2^24
    const int n4      = n_total / 4;          // float4 count (2^22)

    const int block = 256;                    // 8 wave32 waves per block
    const int grid  = 2048;                   // stride 524288 divides n4 -> no tail

    fcb_init<<<1, 32, 0, stream>>>(ws);
    fcb_count<<<grid, block, 0, stream>>>((const float4*)targets, ws, n4);
    fcb_lossred<<<grid, block, 0, stream>>>((const float4*)inputs,
                                            (const float4*)targets, ws,
                                            n4, (float)n_total);
    fcb_finalize<<<1, 32, 0, stream>>>(ws, out, 1.0f / (float)n_total);
}